// node_prompt_layer_feature_cat_edge_21534966022315
// MI455X (gfx1250) — compile-verified
//
#include <hip/hip_runtime.h>
#include <hip/hip_bf16.h>

#define D_FEAT   96
#define P_DIM    96
#define OUT_DIM  (D_FEAT + P_DIM)   // 192

typedef __attribute__((ext_vector_type(2))) float v2f;
typedef __attribute__((ext_vector_type(8))) float v8f;

// ---------------------------------------------------------------------------
// Kernel 1: zero the feature half of out and the degree accumulator (ws).
// One block per node, 96 threads; thread c zeros out[n*192 + c], thread 0
// zeros deg[n].  (Harness poisons d_out/d_ws with 0xAA; must re-zero every
// call for replay determinism.)
// ---------------------------------------------------------------------------
__global__ void zero_kernel(float* __restrict__ out,
                            float* __restrict__ deg,
                            int n_nodes) {
    int n = blockIdx.x;
    if (n >= n_nodes) return;
    int c = threadIdx.x;               // 0..95
    out[(size_t)n * OUT_DIM + c] = 0.0f;
    if (c == 0) deg[n] = 0.0f;
}

// ---------------------------------------------------------------------------
// Kernel 2: edge scatter. One wave32 per edge.
// Edge index is wave-uniform -> readfirstlane so src/dst become scalar loads.
// Lane l handles feature columns l, l+32, l+64: coalesced b32 loads of
// ft[src] and GLOBAL_ATOMIC_ADD_F32 into out[dst] feature half (resolved in
// L2; the 38 MB output and 19 MB gather table both fit in the 192 MB L2).
// Lane 0 accumulates the in-degree with an f32 atomic.
// ---------------------------------------------------------------------------
__global__ void edge_scatter(const float* __restrict__ ft,
                             const int*   __restrict__ src,
                             const int*   __restrict__ dst,
                             float*       __restrict__ out,
                             float*       __restrict__ deg,
                             int n_edges) {
    int gtid = blockIdx.x * blockDim.x + threadIdx.x;
    int e    = __builtin_amdgcn_readfirstlane(gtid >> 5);   // wave-uniform edge id
    int lane = threadIdx.x & 31;
    if (e >= n_edges) return;

    int s = src[e];                    // scalar loads (uniform index)
    int d = dst[e];

    const float* fsrc = ft  + (size_t)s * D_FEAT;
    float*       fdst = out + (size_t)d * OUT_DIM;

#pragma unroll
    for (int k = 0; k < 3; ++k) {
        int c = lane + k * 32;          // 0..95, coalesced per wave
        atomicAdd(fdst + c, fsrc[c]);
    }
    if (lane == 0) {
        atomicAdd(deg + d, 1.0f);
    }
}

// ---------------------------------------------------------------------------
// Kernel 3: prompt half as a rank-1 outer product via V_WMMA_F32_16X16X4_F32.
//   D(16x16) = A(16x4) x B(4x16),  A[:,0] = deg tile, B[0,:] = weight chunk,
//   all other K entries zero  =>  D[m][n] = deg[m] * weight[n] exactly.
// A layout (32-bit A 16x4): lanes 0-15 hold M=0..15; VGPR0 = K=0 (lanes 0-15).
// B layout (32-bit B 4x16): K=0 row lives in VGPR0 lanes 0-15.
// C/D layout: VGPR r: M=r (lanes 0-15) / M=8+r (lanes 16-31), N = lane&15.
// One wave per 16-node tile, 6 column chunks of 16 cover P_DIM=96.
// All lane-dependent masking is done with cndmask (no EXEC divergence), and
// all branches before the WMMA are wave-uniform, so EXEC is all ones at the
// WMMA as the ISA requires.
// ---------------------------------------------------------------------------
__global__ void prompt_wmma(const float* __restrict__ deg,
                            const float* __restrict__ w,     // [96]
                            float*       __restrict__ out,
                            int n_nodes) {
    int gtid = blockIdx.x * blockDim.x + threadIdx.x;
    int wave = gtid >> 5;
    int lane = threadIdx.x & 31;

    int row0 = wave * 16;
    if (row0 >= n_nodes) return;             // wave-uniform exit

    bool full_tile = (row0 + 16 <= n_nodes); // wave-uniform

    // ---- A matrix: deg column in K=0 (VGPR0, lanes 0-15) -------------------
    int r_lo   = row0 + (lane & 15);
    int r_cl   = (r_lo < n_nodes) ? r_lo : (n_nodes - 1);   // in-bounds load
    float dv   = deg[r_cl];                                  // coalesced
    bool a_live = (lane < 16) && (r_lo < n_nodes);
    v2f A;
    A.x = a_live ? dv : 0.0f;                // cndmask, no EXEC change
    A.y = 0.0f;

    int m_base = row0 + ((lane >= 16) ? 8 : 0);
    int n_loc  = lane & 15;

#pragma unroll
    for (int cb = 0; cb < 6; ++cb) {
        // ---- B matrix: weight chunk in K=0 row (VGPR0, lanes 0-15) --------
        float wv = w[cb * 16 + n_loc];       // in-bounds for all lanes
        v2f B;
        B.x = (lane < 16) ? wv : 0.0f;
        B.y = 0.0f;

        v8f C = {};
        C = __builtin_amdgcn_wmma_f32_16x16x4_f32(
                /*neg_a=*/false, A,
                /*neg_b=*/false, B,
                /*c_mod=*/(short)0, C,
                /*reuse_a=*/false, /*reuse_b=*/false);

        int ncol = cb * 16 + n_loc;
        float* p = out + (size_t)m_base * OUT_DIM + D_FEAT + ncol;

        if (full_tile) {                     // wave-uniform: straight-line stores
#pragma unroll
            for (int r = 0; r < 8; ++r) {
                p[(size_t)r * OUT_DIM] = C[r];
            }
        } else {                             // last partial tile only
#pragma unroll
            for (int r = 0; r < 8; ++r) {
                if (m_base + r < n_nodes)
                    p[(size_t)r * OUT_DIM] = C[r];
            }
        }
    }
}

// ---------------------------------------------------------------------------
// Launcher
// ---------------------------------------------------------------------------
extern "C" void kernel_launch(void* const* d_in, const int* in_sizes, int n_in,
                              void* d_out, int out_size, void* d_ws, size_t ws_size,
                              hipStream_t stream) {
    const float* ft  = (const float*)d_in[0];   // [N, 96]
    const float* w   = (const float*)d_in[1];   // [1, 96]
    const int*   src = (const int*)d_in[2];     // [E]
    const int*   dst = (const int*)d_in[3];     // [E]

    const int n_nodes = in_sizes[0] / D_FEAT;
    const int n_edges = in_sizes[2];

    float* out = (float*)d_out;
    float* deg = (float*)d_ws;                  // N floats of scratch

    // 1) zero feature half + degree accumulator
    zero_kernel<<<n_nodes, D_FEAT, 0, stream>>>(out, deg, n_nodes);

    // 2) edge scatter: one wave per edge, 8 waves per 256-thread block
    {
        int blocks = (n_edges + 7) / 8;
        edge_scatter<<<blocks, 256, 0, stream>>>(ft, src, dst, out, deg, n_edges);
    }

    // 3) prompt half: one wave per 16-node tile via WMMA outer product
    {
        int tiles  = (n_nodes + 15) / 16;
        int blocks = (tiles + 7) / 8;           // 8 waves per 256-thread block
        prompt_wmma<<<blocks, 256, 0, stream>>>(deg, w, out, n_nodes);
    }
}